// ManifoldHyperConnection_43258910605336
// MI455X (gfx1250) — compile-verified
//
#include <hip/hip_runtime.h>

typedef __attribute__((ext_vector_type(2))) float v2f;
typedef __attribute__((ext_vector_type(8))) float v8f;

#define KTOT      8192          // reduction length (N*D)
#define NROW      8             // rows of x staged per workgroup
#define ROWSTRIDE 8200          // padded LDS row stride in floats (bank skew, 16B aligned)
#define NWAVE     8             // waves per workgroup (256 threads, wave32)
#define KSLICE    (KTOT / NWAVE)
#define ROWS_TOT  8192          // B*S
#define OUT_PRE   0
#define OUT_POST  (ROWS_TOT * 4)
#define OUT_RES   (ROWS_TOT * 8)

#if __has_builtin(__builtin_amdgcn_global_load_async_to_lds_b128)
#define HAVE_ASYNC_LDS 1
#else
#define HAVE_ASYNC_LDS 0
#endif

// The async-LDS builtin expects pointers to 16-byte int vectors (per hipcc
// diagnostic: "__attribute__((__vector_size__(4 * sizeof(int)))) int *").
typedef int v4i_t __attribute__((vector_size(16)));
typedef __attribute__((address_space(1))) v4i_t GVec4;   // global
typedef __attribute__((address_space(3))) v4i_t LVec4;   // LDS

// ---------------------------------------------------------------------------
// Pack phi_pre (8192x4) | phi_post (8192x4) | phi_res (8192x16) into a
// transposed, zero-padded B matrix phiB[32][8192] so that WMMA B fragments
// are contiguous 8-byte loads per lane.
// ---------------------------------------------------------------------------
__global__ void pack_phi_kernel(const float* __restrict__ phi_pre,
                                const float* __restrict__ phi_post,
                                const float* __restrict__ phi_res,
                                float* __restrict__ phiB) {
  int k = blockIdx.x * blockDim.x + threadIdx.x;
  if (k >= KTOT) return;
#pragma unroll
  for (int n = 0; n < 32; ++n) {
    float v = 0.0f;
    if (n < 4)       v = phi_pre[k * 4 + n];
    else if (n < 8)  v = phi_post[k * 4 + (n - 4)];
    else if (n < 24) v = phi_res[k * 16 + (n - 8)];
    phiB[(size_t)n * KTOT + k] = v;
  }
}

// ---------------------------------------------------------------------------
// Fused RMSNorm + (x_norm @ [phi_pre|phi_post|phi_res]) + sigmoid/sinkhorn.
// One workgroup = 8 rows of x staged in LDS via async Global->LDS DMA
// (single HBM pass over x). 8 waves split K; each wave runs
// V_WMMA_F32_16X16X4_F32 over its K slice for two 16-wide N tiles, then
// reduces via ds_add_f32 atomics in LDS.
// ---------------------------------------------------------------------------
__global__ __launch_bounds__(256)
void manifold_kernel(const float* __restrict__ x,
                     const float* __restrict__ phiB,
                     const float* __restrict__ alpha_pre_p,
                     const float* __restrict__ alpha_post_p,
                     const float* __restrict__ alpha_res_p,
                     const float* __restrict__ b_pre,
                     const float* __restrict__ b_post,
                     const float* __restrict__ b_res,
                     float* __restrict__ out) {
  extern __shared__ float smem[];
  float* xs   = smem;                       // NROW * ROWSTRIDE
  float* cbuf = smem + NROW * ROWSTRIDE;    // 16 x 32 accumulator
  float* rbuf = cbuf + 16 * 32;             // 8 inv-rms values

  const int tid  = threadIdx.x;
  const int wave = tid >> 5;
  const int lane = tid & 31;
  const size_t rowbase = (size_t)blockIdx.x * NROW;

  // ---- Phase 1: stage 8 rows (256KB) into LDS ----
  {
    const float4* gx = (const float4*)(x + rowbase * KTOT);
#pragma unroll
    for (int i = 0; i < (NROW * KTOT / 4) / 256; ++i) {
      int idx = i * 256 + tid;       // float4 index into the 8-row block
      int f   = idx * 4;
      int r   = f >> 13;             // / KTOT
      int c   = f & (KTOT - 1);
#if HAVE_ASYNC_LDS
      // CDNA5 async DMA: global -> LDS without a VGPR round trip (ASYNCcnt)
      __builtin_amdgcn_global_load_async_to_lds_b128(
          (GVec4*)(gx + idx), (LVec4*)(xs + r * ROWSTRIDE + c), 0, 0);
#else
      float4 v = gx[idx];
      *(float4*)(xs + r * ROWSTRIDE + c) = v;
#endif
    }
#if HAVE_ASYNC_LDS
#if __has_builtin(__builtin_amdgcn_s_wait_asynccnt)
    __builtin_amdgcn_s_wait_asynccnt(0);
#else
    asm volatile("s_wait_asynccnt 0x0" ::: "memory");
#endif
#endif
  }
  for (int i = tid; i < 16 * 32; i += 256) cbuf[i] = 0.0f;
  __syncthreads();

  // ---- Phase 1b: per-row sum of squares; wave w owns row w ----
  {
    float ss = 0.0f;
    const float4* row4 = (const float4*)(xs + wave * ROWSTRIDE);
    for (int j = 0; j < KTOT / 128; ++j) {
      float4 v = row4[j * 32 + lane];
      ss = fmaf(v.x, v.x, ss);
      ss = fmaf(v.y, v.y, ss);
      ss = fmaf(v.z, v.z, ss);
      ss = fmaf(v.w, v.w, ss);
    }
#pragma unroll
    for (int off = 16; off > 0; off >>= 1)
      ss += __shfl_xor(ss, off, 32);
    if (lane == 0)
      rbuf[wave] = 1.0f / sqrtf(ss * (1.0f / KTOT) + 1e-6f);
  }
  __syncthreads();

  // ---- Phase 2: WMMA f32 16x16x4; M rows 0..7 valid, N tiles [0..15],[16..31]
  {
    const int m     = lane & 15;                 // M index (rows 8..15 are dead)
    const int khalf = (lane >> 4) << 1;          // lanes 16..31 hold K=2,3
    const float ir  = rbuf[m & 7];               // inv rms for this row

    int kbase = wave * KSLICE + khalf;
    const float* aptr  = xs + m * ROWSTRIDE + kbase;
    const float* b0ptr = phiB + (size_t)m * KTOT + kbase;          // N = m
    const float* b1ptr = phiB + (size_t)(m + 16) * KTOT + kbase;   // N = m+16

    v8f acc0 = {0.f, 0.f, 0.f, 0.f, 0.f, 0.f, 0.f, 0.f};
    v8f acc1 = {0.f, 0.f, 0.f, 0.f, 0.f, 0.f, 0.f, 0.f};

#pragma unroll 4
    for (int it = 0; it < KSLICE / 4; ++it) {
      v2f a = *(const v2f*)aptr;
      // RMSNorm apply: clamp(x * inv_rms, -10, 10)
      a.x = fminf(fmaxf(a.x * ir, -10.0f), 10.0f);
      a.y = fminf(fmaxf(a.y * ir, -10.0f), 10.0f);
      v2f b0 = *(const v2f*)b0ptr;
      v2f b1 = *(const v2f*)b1ptr;
      acc0 = __builtin_amdgcn_wmma_f32_16x16x4_f32(
          false, a, false, b0, (short)0, acc0, false, false);
      acc1 = __builtin_amdgcn_wmma_f32_16x16x4_f32(
          false, a, false, b1, (short)0, acc1, false, false);
      aptr += 4; b0ptr += 4; b1ptr += 4;
    }

    // cross-wave reduction into LDS C buffer (rows 8..15 / cols 24..31 dead)
    const int row0 = (lane >> 4) * 8;
    const int col  = lane & 15;
#pragma unroll
    for (int i = 0; i < 8; ++i) {
      atomicAdd(&cbuf[(row0 + i) * 32 + col],      acc0[i]);
      atomicAdd(&cbuf[(row0 + i) * 32 + 16 + col], acc1[i]);
    }
  }
  __syncthreads();

  // ---- Phase 3: epilogue, one lane per row ----
  if (tid < NROW) {
    const float apre  = alpha_pre_p[0];
    const float apost = alpha_post_p[0];
    const float ares  = alpha_res_p[0];
    const size_t r    = rowbase + tid;
    const float* S    = cbuf + tid * 32;

    float* out_pre  = out + OUT_PRE  + r * 4;
    float* out_post = out + OUT_POST + r * 4;
    float* out_res  = out + OUT_RES  + r * 16;

#pragma unroll
    for (int j = 0; j < 4; ++j) {
      float tp = apre * S[j] + b_pre[j];
      out_pre[j] = 1.0f / (1.0f + __expf(-tp));
      float tq = apost * S[4 + j] + b_post[j];
      out_post[j] = 2.0f / (1.0f + __expf(-tq));
    }

    float M[16];
#pragma unroll
    for (int i = 0; i < 16; ++i) {
      float t = ares * S[8 + i] + b_res[i];
      t = fminf(fmaxf(t, -20.0f), 20.0f);
      M[i] = __expf(t);
    }
    for (int itn = 0; itn < 20; ++itn) {
#pragma unroll
      for (int i = 0; i < 4; ++i) {
        float s = M[i*4] + M[i*4+1] + M[i*4+2] + M[i*4+3] + 1e-6f;
        float inv = 1.0f / s;
        M[i*4] *= inv; M[i*4+1] *= inv; M[i*4+2] *= inv; M[i*4+3] *= inv;
      }
#pragma unroll
      for (int j = 0; j < 4; ++j) {
        float s = M[j] + M[4+j] + M[8+j] + M[12+j] + 1e-6f;
        float inv = 1.0f / s;
        M[j] *= inv; M[4+j] *= inv; M[8+j] *= inv; M[12+j] *= inv;
      }
    }
#pragma unroll
    for (int i = 0; i < 16; ++i) out_res[i] = M[i];
  }
}

extern "C" void kernel_launch(void* const* d_in, const int* in_sizes, int n_in,
                              void* d_out, int out_size, void* d_ws, size_t ws_size,
                              hipStream_t stream) {
  (void)in_sizes; (void)n_in; (void)out_size; (void)ws_size;
  const float* x          = (const float*)d_in[0];
  const float* phi_pre    = (const float*)d_in[1];
  const float* phi_post   = (const float*)d_in[2];
  const float* phi_res    = (const float*)d_in[3];
  const float* alpha_pre  = (const float*)d_in[4];
  const float* alpha_post = (const float*)d_in[5];
  const float* alpha_res  = (const float*)d_in[6];
  const float* b_pre      = (const float*)d_in[7];
  const float* b_post     = (const float*)d_in[8];
  const float* b_res      = (const float*)d_in[9];
  float* out  = (float*)d_out;
  float* phiB = (float*)d_ws;   // 32 * 8192 floats = 1 MB

  pack_phi_kernel<<<KTOT / 256, 256, 0, stream>>>(phi_pre, phi_post, phi_res, phiB);

  size_t smem = (size_t)(NROW * ROWSTRIDE + 16 * 32 + 16) * sizeof(float);
  manifold_kernel<<<ROWS_TOT / NROW, 256, smem, stream>>>(
      x, phiB, alpha_pre, alpha_post, alpha_res, b_pre, b_post, b_res, out);
}